// MambaFilter_90950227460137
// MI455X (gfx1250) — compile-verified
//
#include <hip/hip_runtime.h>

typedef __attribute__((ext_vector_type(2))) float v2f;
typedef __attribute__((ext_vector_type(8))) float v8f;

#define D_STATE 4096
#define D_CONV  16
#define T_IN    (D_STATE - D_CONV + 1)   // 4081
#define PAD     (D_CONV - 1)             // 15

// ws float-index layout (all regions written each launch; no persistent state)
#define WS_WTIL 0      // [16]  w~ = conv_w^T . proj_w
#define WS_BT   16     // bt   = conv_b . proj_w
#define WS_ASUM 17     // a    = sum(exp(A_p*time_decay))
#define WS_AVEC 64     // [4096] A[s]
#define WS_Y    8192   // [4096] y[t] = xc[t,:] . proj_w

// --- Kernel 1: A[s] = exp(A_p*td), and a = sum(A) -------------------------
__global__ void k_decay(const float* __restrict__ A_p,
                        const float* __restrict__ td,
                        float* __restrict__ ws) {
  __shared__ float red[256];
  const int tid = threadIdx.x;
  float partial = 0.f;
  for (int j = 0; j < D_STATE / 256; ++j) {
    const int s = tid + 256 * j;
    const float a = expf(A_p[s] * td[s]);
    ws[WS_AVEC + s] = a;
    partial += a;
  }
  red[tid] = partial;
  __syncthreads();
  for (int off = 128; off > 0; off >>= 1) {
    if (tid < off) red[tid] += red[tid + off];
    __syncthreads();
  }
  if (tid == 0) ws[WS_ASUM] = red[0];
}

// --- Kernel 2: w~[k] = sum_c conv_w[c,k]*proj_w[c] via f32 WMMA; bt -------
// One block, 8 waves; wave w reduces channels [w*512, w*512+512) in K=4 chunks.
// A-frag (16x4 f32): lanes 0-15 hold {K=0,K=1}, lanes 16-31 hold {K=2,K=3},
// M = tap index. B-frag rows = proj_w chunk, replicated over N.
__global__ void k_wtil(const float* __restrict__ conv_w,   // [4096][16] row-major
                       const float* __restrict__ conv_b,
                       const float* __restrict__ proj_w,
                       float* __restrict__ ws) {
  __shared__ float wpart[8][16];
  __shared__ float red[256];
  const int tid  = threadIdx.x;
  const int wave = tid >> 5;
  const int lane = tid & 31;
  const int m    = lane & 15;
  const bool hi  = lane >= 16;

  v8f acc = {};
  const int c_beg = wave * (D_STATE / 8);
  const int c_end = c_beg + (D_STATE / 8);
  for (int c0 = c_beg; c0 < c_end; c0 += 4) {
    const int cA = c0 + (hi ? 2 : 0);
    v2f a, b;
    a.x = conv_w[(cA + 0) * D_CONV + m];
    a.y = conv_w[(cA + 1) * D_CONV + m];
    b.x = proj_w[cA + 0];
    b.y = proj_w[cA + 1];
    acc = __builtin_amdgcn_wmma_f32_16x16x4_f32(false, a, false, b,
                                                (short)0, acc, false, false);
  }
  // C/D layout: VGPR r -> M=r (lanes 0-15, N=lane), M=8+r (lanes 16-31)
  if (lane == 0)
    for (int r = 0; r < 8; ++r) wpart[wave][r] = acc[r];
  if (lane == 16)
    for (int r = 0; r < 8; ++r) wpart[wave][8 + r] = acc[r];

  float p = 0.f;
  for (int j = 0; j < D_STATE / 256; ++j) {
    const int c = tid + 256 * j;
    p += conv_b[c] * proj_w[c];
  }
  red[tid] = p;
  __syncthreads();
  for (int off = 128; off > 0; off >>= 1) {
    if (tid < off) red[tid] += red[tid + off];
    __syncthreads();
  }
  if (tid == 0) ws[WS_BT] = red[0];
  if (tid < 16) {
    float s = 0.f;
    for (int w = 0; w < 8; ++w) s += wpart[w][tid];
    ws[WS_WTIL + tid] = s;
  }
}

// --- Kernel 3: y = X_band(4096x16) @ w~ via 4 K=4 WMMAs per 16-row tile ---
__global__ void k_fir(const float* __restrict__ x,
                      float* __restrict__ ws) {
  const int tid  = threadIdx.x;
  const int wave = tid >> 5;
  const int lane = tid & 31;
  const int m    = lane & 15;
  const bool hi  = lane >= 16;
  const int t0   = (blockIdx.x * 8 + wave) * 16;

  const float bt = ws[WS_BT];
  v8f acc = {};
#pragma unroll
  for (int q = 0; q < 4; ++q) {
    const int j0 = q * 4 + (hi ? 2 : 0);      // tap index for this half-wave
    const int i0 = t0 + m - PAD + j0;
    const int i1 = i0 + 1;
    v2f a, b;
    a.x = (i0 >= 0 && i0 < T_IN) ? x[i0] : 0.f;
    a.y = (i1 >= 0 && i1 < T_IN) ? x[i1] : 0.f;
    b.x = ws[WS_WTIL + j0];
    b.y = ws[WS_WTIL + j0 + 1];
    acc = __builtin_amdgcn_wmma_f32_16x16x4_f32(false, a, false, b,
                                                (short)0, acc, false, false);
  }
  if (lane == 0)
    for (int r = 0; r < 8; ++r) ws[WS_Y + t0 + r] = acc[r] + bt;
  if (lane == 16)
    for (int r = 0; r < 8; ++r) ws[WS_Y + t0 + 8 + r] = acc[r] + bt;
}

// --- Kernel 4: u0p = A.y ; G via scalar geometric scan ; write output -----
__global__ void k_final(const float* __restrict__ proj_b,
                        const float* __restrict__ ws,
                        float* __restrict__ out) {
  __shared__ float red[1024];
  __shared__ float s_add;
  const int tid = threadIdx.x;
  float p = 0.f;
  for (int j = 0; j < 4; ++j) {
    const int s = tid + 1024 * j;
    p += ws[WS_AVEC + s] * ws[WS_Y + s];
  }
  red[tid] = p;
  __syncthreads();
  for (int off = 512; off > 0; off >>= 1) {
    if (tid < off) red[tid] += red[tid + off];
    __syncthreads();
  }
  if (tid == 0) {
    const float u0p = red[0];
    const float a = ws[WS_ASUM];
    float G = 0.f, g = 1.f;                  // u_k = u0*g_k ; acc = u0*G
    for (int k = 0; k < D_STATE; ++k) { G += g; g = fmaf(a, g, 1.f); }
    s_add = fmaf(G, u0p, proj_b[0]);
  }
  __syncthreads();
  const float add = s_add;
  for (int j = 0; j < 4; ++j) {
    const int t = tid + 1024 * j;
    out[t] = fmaf((float)(D_STATE + 1), ws[WS_Y + t], add);
  }
}

extern "C" void kernel_launch(void* const* d_in, const int* in_sizes, int n_in,
                              void* d_out, int out_size, void* d_ws, size_t ws_size,
                              hipStream_t stream) {
  const float* x      = (const float*)d_in[0];
  const float* conv_w = (const float*)d_in[1];
  const float* conv_b = (const float*)d_in[2];
  const float* A_p    = (const float*)d_in[3];
  const float* td     = (const float*)d_in[4];
  const float* proj_w = (const float*)d_in[5];
  const float* proj_b = (const float*)d_in[6];
  float* ws  = (float*)d_ws;
  float* out = (float*)d_out;

  k_decay<<<1, 256, 0, stream>>>(A_p, td, ws);
  k_wtil <<<1, 256, 0, stream>>>(conv_w, conv_b, proj_w, ws);
  k_fir  <<<32, 256, 0, stream>>>(x, ws);
  k_final<<<1, 1024, 0, stream>>>(proj_b, ws, out);
}